// MetricLearningLoss1_53197464928782
// MI455X (gfx1250) — compile-verified
//
#include <hip/hip_runtime.h>
#include <hip/hip_bf16.h>

typedef float v2f __attribute__((ext_vector_type(2)));
typedef float v8f __attribute__((ext_vector_type(8)));

#define DK 256           // embedding dim
#define LDSS 260         // padded LDS row stride (floats) -> conflict-free banks
#define MARGIN 0.075f

// ---------------------------------------------------------------------------
// K1: exact top-k selection by rank counting (reproduces jax.lax.top_k set,
// ties broken by smaller index). O(N^2) compares, conf fully L2-resident.
// ---------------------------------------------------------------------------
__global__ __launch_bounds__(256) void rank_select_kernel(
    const float* __restrict__ conf, int N, int K, int* __restrict__ sel) {
  __shared__ float sc[256];
  const int i = blockIdx.x * 256 + threadIdx.x;
  const float mc = conf[i];
  int rank = 0;
  for (int t = 0; t < N; t += 256) {
    sc[threadIdx.x] = conf[t + threadIdx.x];
    __syncthreads();
#pragma unroll 8
    for (int j2 = 0; j2 < 256; ++j2) {
      const float c2 = sc[j2];
      const int j = t + j2;
      rank += (c2 > mc) || ((c2 == mc) && (j < i));
    }
    __syncthreads();
  }
  if (rank < K) sel[rank] = i;
}

// ---------------------------------------------------------------------------
// K2: gather selected rows, compute squared norms, copy labels; zero-pad rows
// K..Kpad so WMMA tiles are full.
// ---------------------------------------------------------------------------
__global__ __launch_bounds__(64) void gather_kernel(
    const float* __restrict__ emb, const int* __restrict__ tags,
    const int* __restrict__ sel, int K,
    float* __restrict__ E, float* __restrict__ sqn, int* __restrict__ labs) {
  const int r = blockIdx.x;
  const int t = threadIdx.x;
  __shared__ float red[64];
  float acc = 0.0f;
  float* d = E + (long)r * DK;
  if (r < K) {
    const float* s = emb + (long)sel[r] * DK;
    for (int c = t; c < DK; c += 64) {
      const float v = s[c];
      d[c] = v;
      acc += v * v;
    }
  } else {
    for (int c = t; c < DK; c += 64) d[c] = 0.0f;
  }
  red[t] = acc;
  __syncthreads();
  for (int off = 32; off > 0; off >>= 1) {
    if (t < off) red[t] += red[t + off];
    __syncthreads();
  }
  if (t == 0) {
    sqn[r] = red[0];
    labs[r] = (r < K) ? tags[sel[r]] : -1;
  }
}

// ---------------------------------------------------------------------------
// K3: fused Gram-matrix + hard-mining. One block = 16 anchor rows.
// 4 waves sweep all column tiles twice (pass1: hard_pos / neg_min,
// pass2: semi-hard-neg min using pass1's per-row hard_pos from LDS).
// Inner product tiles computed with V_WMMA_F32_16X16X4_F32.
// Masking epilogue is fully branchless (cndmask-fed max/min), so no EXEC
// juggling and the wave stays convergent between WMMA tile loops.
// ---------------------------------------------------------------------------
__global__ __launch_bounds__(128) void triplet_kernel(
    const float* __restrict__ E, const float* __restrict__ sqn,
    const int* __restrict__ labs, int K, int NT,
    float* __restrict__ pl, float* __restrict__ pc) {
  __shared__ float Alds[16 * LDSS];
  __shared__ float cmb[3][4][16];
  __shared__ float hpF[16], nmF[16];
  __shared__ float perA[16], perC[16];

  const int tid  = threadIdx.x;
  const int wave = tid >> 5;
  const int lane = tid & 31;
  const int l16  = lane & 15;
  const int hi   = lane >> 4;
  const int m0   = blockIdx.x * 16;
  const float INF = __builtin_inff();

  // Stage A tile (16 rows x 256) into LDS with padded stride.
  for (int idx = tid; idx < 16 * DK; idx += 128) {
    const int r = idx >> 8;
    const int c = idx & (DK - 1);
    Alds[r * LDSS + c] = E[(long)(m0 + r) * DK + c];
  }
  __syncthreads();

  // Per-lane row metadata for the 8 C-matrix slots (ISA C-layout: M = s + 8*hi).
  float sqr[8];
  int   labr[8], rowg[8];
#pragma unroll
  for (int s = 0; s < 8; ++s) {
    const int r = m0 + s + 8 * hi;
    rowg[s] = r;
    sqr[s]  = sqn[r];
    labr[s] = labs[r];
  }

  float rp[8], rn[8];
#pragma unroll
  for (int s = 0; s < 8; ++s) { rp[s] = -1.0f; rn[s] = INF; }

  const float* Ap = Alds + l16 * LDSS + 2 * hi;

  // ---------------- PASS 1: hard positive (max) + negative min ------------
  for (int nt = wave; nt < NT; nt += 4) {
    const int j = nt * 16 + l16;
    const int   labj = labs[j];
    const float sqj  = sqn[j];
    const bool  jv   = (j < K);
    const float* Bp  = E + (long)j * DK + 2 * hi;
    v8f c = {};
#pragma unroll 4
    for (int k0 = 0; k0 < DK; k0 += 4) {
      v2f a = *(const v2f*)(Ap + k0);
      v2f b = *(const v2f*)(Bp + k0);
      c = __builtin_amdgcn_wmma_f32_16x16x4_f32(false, a, false, b,
                                                (short)0, c, false, false);
    }
#pragma unroll
    for (int s = 0; s < 8; ++s) {
      const float sq   = fmaxf(sqr[s] + sqj - 2.0f * c[s], 0.0f);
      const bool  same = (labr[s] == labj);
      const bool  posm = jv & same & (j != rowg[s]);
      const bool  negm = jv & !same;
      rp[s] = fmaxf(rp[s], posm ? sq : -1.0f);   // branchless: cndmask + max
      rn[s] = fminf(rn[s], negm ? sq : INF);     // branchless: cndmask + min
    }
  }
  // reduce across the 16 lanes of each half (halves hold disjoint row sets)
#pragma unroll
  for (int m = 1; m <= 8; m <<= 1) {
#pragma unroll
    for (int s = 0; s < 8; ++s) {
      rp[s] = fmaxf(rp[s], __shfl_xor(rp[s], m, 32));
      rn[s] = fminf(rn[s], __shfl_xor(rn[s], m, 32));
    }
  }
  if (l16 == 0) {
#pragma unroll
    for (int s = 0; s < 8; ++s) {
      cmb[0][wave][s + 8 * hi] = rp[s];
      cmb[1][wave][s + 8 * hi] = rn[s];
    }
  }
  __syncthreads();
  if (tid < 16) {
    hpF[tid] = fmaxf(fmaxf(cmb[0][0][tid], cmb[0][1][tid]),
                     fmaxf(cmb[0][2][tid], cmb[0][3][tid]));
    nmF[tid] = fminf(fminf(cmb[1][0][tid], cmb[1][1][tid]),
                     fminf(cmb[1][2][tid], cmb[1][3][tid]));
  }
  __syncthreads();

  float hpr[8], rs[8];
#pragma unroll
  for (int s = 0; s < 8; ++s) { hpr[s] = hpF[s + 8 * hi]; rs[s] = INF; }

  // ---------------- PASS 2: semi-hard negative min ------------------------
  for (int nt = wave; nt < NT; nt += 4) {
    const int j = nt * 16 + l16;
    const int   labj = labs[j];
    const float sqj  = sqn[j];
    const bool  jv   = (j < K);
    const float* Bp  = E + (long)j * DK + 2 * hi;
    v8f c = {};
#pragma unroll 4
    for (int k0 = 0; k0 < DK; k0 += 4) {
      v2f a = *(const v2f*)(Ap + k0);
      v2f b = *(const v2f*)(Bp + k0);
      c = __builtin_amdgcn_wmma_f32_16x16x4_f32(false, a, false, b,
                                                (short)0, c, false, false);
    }
#pragma unroll
    for (int s = 0; s < 8; ++s) {
      const float sq  = fmaxf(sqr[s] + sqj - 2.0f * c[s], 0.0f);
      const bool  shm = jv & (labr[s] != labj) & (sq > hpr[s]);
      rs[s] = fminf(rs[s], shm ? sq : INF);      // branchless
    }
  }
#pragma unroll
  for (int m = 1; m <= 8; m <<= 1) {
#pragma unroll
    for (int s = 0; s < 8; ++s) rs[s] = fminf(rs[s], __shfl_xor(rs[s], m, 32));
  }
  if (l16 == 0) {
#pragma unroll
    for (int s = 0; s < 8; ++s) cmb[2][wave][s + 8 * hi] = rs[s];
  }
  __syncthreads();

  // ---------------- finalize 16 anchors of this block ---------------------
  if (tid < 16) {
    const int r = m0 + tid;
    float res = 0.0f, cv = 0.0f;
    if (r < K) {
      const float hp = hpF[tid];
      const float nm = nmF[tid];
      const float sh = fminf(fminf(cmb[2][0][tid], cmb[2][1][tid]),
                             fminf(cmb[2][2][tid], cmb[2][3][tid]));
      const bool valid = (hp >= 0.0f) && (nm < INF);
      const float hn = (sh < INF) ? sh : nm;
      if (valid) {
        res = fmaxf(sqrtf(hp) - sqrtf(hn) + MARGIN, 0.0f);
        cv = 1.0f;
      }
    }
    perA[tid] = res;
    perC[tid] = cv;
  }
  __syncthreads();
  if (tid == 0) {
    float a = 0.0f, b = 0.0f;
#pragma unroll
    for (int s = 0; s < 16; ++s) { a += perA[s]; b += perC[s]; }
    pl[blockIdx.x] = a;
    pc[blockIdx.x] = b;
  }
}

// ---------------------------------------------------------------------------
// K4: deterministic final reduction -> scalar loss
// ---------------------------------------------------------------------------
__global__ __launch_bounds__(512) void final_reduce_kernel(
    const float* __restrict__ pl, const float* __restrict__ pc, int nblk,
    float* __restrict__ out) {
  __shared__ float s1[512], s2[512];
  const int t = threadIdx.x;
  float a = 0.0f, b = 0.0f;
  for (int i = t; i < nblk; i += 512) { a += pl[i]; b += pc[i]; }
  s1[t] = a; s2[t] = b;
  __syncthreads();
  for (int off = 256; off > 0; off >>= 1) {
    if (t < off) { s1[t] += s1[t + off]; s2[t] += s2[t + off]; }
    __syncthreads();
  }
  if (t == 0) out[0] = (s2[0] > 0.0f) ? s1[0] / s2[0] : 0.0f;
}

// ---------------------------------------------------------------------------
extern "C" void kernel_launch(void* const* d_in, const int* in_sizes, int n_in,
                              void* d_out, int out_size, void* d_ws, size_t ws_size,
                              hipStream_t stream) {
  const float* emb  = (const float*)d_in[0];
  const int*   tags = (const int*)d_in[1];
  const float* conf = (const float*)d_in[2];

  const int N    = in_sizes[2];          // 32768
  const int K    = N / 5;                // int(0.2 * N) = 6553
  const int Kpad = (K + 15) & ~15;       // 6560
  const int NT   = Kpad / 16;            // column tiles / row blocks

  // workspace carve-up (~6.8 MB)
  float* E    = (float*)d_ws;                    // Kpad * DK
  float* sqn  = E + (size_t)Kpad * DK;           // Kpad
  int*   labs = (int*)(sqn + Kpad);              // Kpad
  int*   sel  = labs + Kpad;                     // Kpad
  float* pl   = (float*)(sel + Kpad);            // NT
  float* pc   = pl + NT;                         // NT

  rank_select_kernel<<<N / 256, 256, 0, stream>>>(conf, N, K, sel);
  gather_kernel<<<Kpad, 64, 0, stream>>>(emb, tags, sel, K, E, sqn, labs);
  triplet_kernel<<<NT, 128, 0, stream>>>(E, sqn, labs, K, NT, pl, pc);
  final_reduce_kernel<<<1, 512, 0, stream>>>(pl, pc, NT, (float*)d_out);
}